// RadialFunctionT_35631048687738
// MI455X (gfx1250) — compile-verified
//
#include <hip/hip_runtime.h>
#include <stdint.h>

namespace {
constexpr int   kNB    = 7;
constexpr int   kNR    = 5;
constexpr int   kNS    = 119;
constexpr int   kNPair = kNS * kNS;          // 14161
constexpr int   kPadW  = 36;                 // 35 coeffs + 1 pad -> 144B, 16B aligned
constexpr int   kTPB   = 128;                // 4 waves/block
constexpr int   kEPT   = 4;                  // edges per thread (pipelined)
constexpr float kBetta = 1.36111111f;        // 49/36
constexpr float kDelta = 0.785714286f;       // 5.5/7
constexpr float kRMin  = 0.5f;
constexpr float kRMax  = 6.0f;
constexpr float kRadNorm   = 0.96480453f;    // (2*BETTA/pi)^0.25
constexpr float kEmbNorm   = 0.37796447f;    // 1/sqrt(7)
constexpr float kBothNorm  = 0.36466190f;    // kRadNorm * kEmbNorm
constexpr float kPiOverRmx = 0.52359878f;    // pi/6
}

// ---------------------------------------------------------------------------
// gfx1250 async gather: per-lane global address -> per-lane LDS slot.
// Tracked with ASYNCcnt; completion enforced with s_wait_asynccnt.
// INST_OFFSET is added to BOTH the LDS address and the memory address
// (CDNA5 ISA), so one (lds, gaddr) pair covers all 9 b128s.
// ---------------------------------------------------------------------------
#define ASYNC_LD_B128(LDSA, GA, OFF)                                        \
  asm volatile("global_load_async_to_lds_b128 %0, %1, off offset:" OFF      \
               :: "v"(LDSA), "v"(GA) : "memory")

#define ASYNC_GATHER_144B(LDSA, GA) do {                                    \
  ASYNC_LD_B128(LDSA, GA, "0");   ASYNC_LD_B128(LDSA, GA, "16");            \
  ASYNC_LD_B128(LDSA, GA, "32");  ASYNC_LD_B128(LDSA, GA, "48");            \
  ASYNC_LD_B128(LDSA, GA, "64");  ASYNC_LD_B128(LDSA, GA, "80");            \
  ASYNC_LD_B128(LDSA, GA, "96");  ASYNC_LD_B128(LDSA, GA, "112");           \
  ASYNC_LD_B128(LDSA, GA, "128");                                           \
} while (0)

// ---------------------------------------------------------------------------
// Prep: pad [14161][35] -> [14161][36] floats (144B rows, 16B aligned),
// with EMBED_NORM pre-folded. Table is ~2MB -> lives in L2 (192MB) and is
// the one data structure we want to keep temporally cached.
// ---------------------------------------------------------------------------
__global__ __launch_bounds__(256) void prep_kernel(const float* __restrict__ emb,
                                                   float* __restrict__ ptab) {
  int i = blockIdx.x * 256 + threadIdx.x;
  if (i >= kNPair * kPadW) return;
  int p = i / kPadW;
  int k = i - p * kPadW;
  ptab[i] = (k < 35) ? kEmbNorm * emb[p * 35 + k] : 0.0f;
}

// ---------------------------------------------------------------------------
// Main kernel: 1 thread = 1 edge per iteration, kEPT iterations, with the
// 144B coefficient gather staged through LDS via async loads (double buffer).
// Slot stride = 36 dwords: lanes 0..15 hit all 64 banks exactly once per
// ds_load_b128 pass -> conflict-optimal. Streaming inputs/outputs use
// non-temporal hints so L2 capacity stays reserved for the gather table.
// ---------------------------------------------------------------------------
__global__ __launch_bounds__(kTPB) void radial_async_kernel(
    const float* __restrict__ dr, const int* __restrict__ zi,
    const int* __restrict__ zj, const float* __restrict__ ptab,
    float* __restrict__ out, int n)
{
  __shared__ __align__(16) float smem[2][kTPB * kPadW];
  const int tid = threadIdx.x;
  const long base = (long)blockIdx.x * (kTPB * kEPT) + tid;

  // Preload all per-iteration scalars (coalesced, read-once -> NT) and
  // gather base pointers.
  long e[kEPT]; float d[kEPT]; const float* g[kEPT];
#pragma unroll
  for (int i = 0; i < kEPT; ++i) {
    long ee = base + (long)i * kTPB;
    e[i] = ee;
    long ec = (ee < n) ? ee : (long)(n - 1);   // clamp: keep all lanes active
    d[i] = __builtin_nontemporal_load(dr + ec);
    int p = __builtin_nontemporal_load(zj + ec) * kNS +
            __builtin_nontemporal_load(zi + ec);
    g[i] = ptab + (long)p * kPadW;
  }

  unsigned lds0 = (unsigned)(uintptr_t)&smem[0][tid * kPadW];
  unsigned lds1 = (unsigned)(uintptr_t)&smem[1][tid * kPadW];

  // Prologue: issue gather for iteration 0.
  ASYNC_GATHER_144B(lds0, g[0]);

#pragma unroll
  for (int i = 0; i < kEPT; ++i) {
    if (i + 1 < kEPT) {
      // Issue next tile's gather into the other buffer, then wait for ours:
      // async loads complete in order, so cnt<=9 means iteration i is done.
      ASYNC_GATHER_144B(((i + 1) & 1) ? lds1 : lds0, g[i + 1]);
      asm volatile("s_wait_asynccnt 0x9" ::: "memory");
    } else {
      asm volatile("s_wait_asynccnt 0x0" ::: "memory");
    }

    // Read our 144B slot back as 9x ds_load_b128.
    const float* s = &smem[i & 1][tid * kPadW];
    float cf[36];
#pragma unroll
    for (int k = 0; k < 9; ++k) {
      float4 v = reinterpret_cast<const float4*>(s)[k];
      cf[4 * k + 0] = v.x; cf[4 * k + 1] = v.y;
      cf[4 * k + 2] = v.z; cf[4 * k + 3] = v.w;
    }

    const float dd = d[i];
    float basis[kNB];
#pragma unroll
    for (int b = 0; b < kNB; ++b) {
      float t = (kRMin + kDelta * (float)b) - dd;
      basis[b] = kRadNorm * __expf(-kBetta * t * t);   // v_exp_f32
    }
    const float dc  = fminf(dd, kRMax);
    const float cut = 0.5f * __cosf(kPiOverRmx * dc) + 0.5f;  // v_cos_f32

    if (e[i] < n) {
      float* o = out + (size_t)e[i] * kNR;
#pragma unroll
      for (int r = 0; r < kNR; ++r) {
        float acc = 0.0f;
#pragma unroll
        for (int b = 0; b < kNB; ++b)
          acc = fmaf(cf[r * kNB + b], basis[b], acc);
        __builtin_nontemporal_store(acc * cut, o + r);  // write-once stream
      }
    }
  }
}

// ---------------------------------------------------------------------------
// Fallback (only if d_ws is too small for the padded table): direct gather.
// ---------------------------------------------------------------------------
__global__ __launch_bounds__(256) void radial_direct_kernel(
    const float* __restrict__ dr, const int* __restrict__ zi,
    const int* __restrict__ zj, const float* __restrict__ emb,
    float* __restrict__ out, int n)
{
  int i = blockIdx.x * 256 + threadIdx.x;
  if (i >= n) return;
  const float dd = __builtin_nontemporal_load(dr + i);
  const int p = __builtin_nontemporal_load(zj + i) * kNS +
                __builtin_nontemporal_load(zi + i);
  const float* cf = emb + (long)p * 35;

  float basis[kNB];
#pragma unroll
  for (int b = 0; b < kNB; ++b) {
    float t = (kRMin + kDelta * (float)b) - dd;
    basis[b] = kBothNorm * __expf(-kBetta * t * t);
  }
  const float dc  = fminf(dd, kRMax);
  const float cut = 0.5f * __cosf(kPiOverRmx * dc) + 0.5f;

  float* o = out + (size_t)i * kNR;
#pragma unroll
  for (int r = 0; r < kNR; ++r) {
    float acc = 0.0f;
#pragma unroll
    for (int b = 0; b < kNB; ++b)
      acc = fmaf(cf[r * kNB + b], basis[b], acc);
    __builtin_nontemporal_store(acc * cut, o + r);
  }
}

extern "C" void kernel_launch(void* const* d_in, const int* in_sizes, int n_in,
                              void* d_out, int out_size, void* d_ws, size_t ws_size,
                              hipStream_t stream) {
  const float* dr  = (const float*)d_in[0];
  const int*   zi  = (const int*)d_in[1];
  const int*   zj  = (const int*)d_in[2];
  const float* emb = (const float*)d_in[3];
  float* out = (float*)d_out;
  const int n = in_sizes[0];
  if (n <= 0) return;

  const size_t need = (size_t)kNPair * kPadW * sizeof(float);  // ~2.04 MB
  if (ws_size >= need) {
    float* ptab = (float*)d_ws;
    const int tot = kNPair * kPadW;
    prep_kernel<<<(tot + 255) / 256, 256, 0, stream>>>(emb, ptab);
    const int blocks = (n + kTPB * kEPT - 1) / (kTPB * kEPT);
    radial_async_kernel<<<blocks, kTPB, 0, stream>>>(dr, zi, zj, ptab, out, n);
  } else {
    radial_direct_kernel<<<(n + 255) / 256, 256, 0, stream>>>(dr, zi, zj, emb, out, n);
  }
}